// SpconvBackbone_18202071400551
// MI455X (gfx1250) — compile-verified
//
#include <hip/hip_runtime.h>

// ---------------------------------------------------------------------------
// SpconvBackbone for MI455X (gfx1250, wave32).
// Conv GEMMs run on V_WMMA_F32_16X16X4_F32 (full f32 precision, matches ref).
// Weights pre-swizzled into per-wave B fragments; 4 voxel tiles per wave to
// amortize B traffic (VMEM:WMMA ~1.3:1).
// ---------------------------------------------------------------------------

static constexpr int kB    = 8;
static constexpr int kN    = 65536;
static constexpr int kBN   = kB * kN;          // 524288 voxel rows / points
static constexpr int kSENT = 2000000000;
#define kVS  0.4f
#define kEPS 1e-5f

typedef __attribute__((ext_vector_type(2))) float v2f;
typedef __attribute__((ext_vector_type(8))) float v8f;

// ---------------- workspace byte offsets ----------------
static constexpr size_t OFF_ORIGIN = 0;                                  // 3 x u32 (float bits)
static constexpr size_t OFF_NVALID = 16;                                 // 1 x i32
static constexpr size_t OFF_BNSUM  = 256;                                // 128 x f32
static constexpr size_t OFF_BNSQ   = 1024;                               // 128 x f32
static constexpr size_t OFF_WT     = 4096;                               // swizzled weights (<=884736 B)
static constexpr size_t OFF_CODES  = OFF_WT    + (size_t)1048576;
static constexpr size_t OFF_SORTA  = OFF_CODES + (size_t)kBN * 4;
static constexpr size_t OFF_SORTB  = OFF_SORTA + (size_t)kBN * 4;
static constexpr size_t OFF_UNQ    = OFF_SORTB + (size_t)kBN * 4;
static constexpr size_t OFF_INV    = OFF_UNQ   + (size_t)kBN * 4;
static constexpr size_t OFF_CNT    = OFF_INV   + (size_t)kBN * 4;
static constexpr size_t OFF_NBR    = OFF_CNT   + (size_t)kBN * 4;        // (BN,27) i32
static constexpr size_t OFF_VFEAT  = OFF_NBR   + (size_t)kBN * 27 * 4;   // (BN,16) f32 (also fsum)
static constexpr size_t OFF_H1     = OFF_VFEAT + (size_t)kBN * 16 * 4;   // (BN,32)
static constexpr size_t OFF_H2     = OFF_H1    + (size_t)kBN * 32 * 4;   // (BN,64)
static constexpr size_t OFF_H3     = OFF_H2    + (size_t)kBN * 64 * 4;   // (BN,128)

// ---------------- init ----------------
__global__ void k_init(unsigned* origin, int* nvalid) {
  if (threadIdx.x < 3) origin[threadIdx.x] = 0x7F7FFFFFu;  // +FLT_MAX bits
  if (threadIdx.x == 3) *nvalid = 0;
}

// global min of xyz per coordinate (all values >= 0 -> uint-bit compare works)
__global__ void k_min_xyz(const float* __restrict__ xyz, unsigned* __restrict__ originBits) {
  __shared__ unsigned smin[3];
  if (threadIdx.x < 3) smin[threadIdx.x] = 0x7F7FFFFFu;
  __syncthreads();
  unsigned l0 = 0x7F7FFFFFu, l1 = 0x7F7FFFFFu, l2 = 0x7F7FFFFFu;
  const long total = (long)kBN * 3;
  for (long i = blockIdx.x * (long)blockDim.x + threadIdx.x; i < total;
       i += (long)gridDim.x * blockDim.x) {
    unsigned bits = __float_as_uint(xyz[i]);
    int c = (int)(i % 3);
    if (c == 0) { if (bits < l0) l0 = bits; }
    else if (c == 1) { if (bits < l1) l1 = bits; }
    else { if (bits < l2) l2 = bits; }
  }
  atomicMin(&smin[0], l0); atomicMin(&smin[1], l1); atomicMin(&smin[2], l2);
  __syncthreads();
  if (threadIdx.x < 3) atomicMin(&originBits[threadIdx.x], smin[threadIdx.x]);
}

// voxel codes per point
__global__ void k_codes(const float* __restrict__ xyz, const unsigned* __restrict__ originBits,
                        int* __restrict__ codes, int* __restrict__ sortA) {
  int p = blockIdx.x * blockDim.x + threadIdx.x;
  if (p >= kBN) return;
  float ox = __uint_as_float(originBits[0]);
  float oy = __uint_as_float(originBits[1]);
  float oz = __uint_as_float(originBits[2]);
  float x = xyz[p * 3 + 0], y = xyz[p * 3 + 1], z = xyz[p * 3 + 2];
  int vx = (int)((x - ox) / kVS); if (vx < 0) vx = 0;
  int vy = (int)((y - oy) / kVS); if (vy < 0) vy = 0;
  int vz = (int)((z - oz) / kVS); if (vz < 0) vz = 0;
  int lin = vx + 1000 * (vy + 1000 * vz);
  codes[p] = lin;
  sortA[p] = lin;
}

// unq := SENT, cnt := 0, fsum := 0
__global__ void k_prep(int* __restrict__ unq, float* __restrict__ cnt, float* __restrict__ fsum) {
  long i = blockIdx.x * (long)blockDim.x + threadIdx.x;
  if (i < (long)kBN * 16) fsum[i] = 0.f;
  if (i < kBN) { unq[i] = kSENT; cnt[i] = 0.f; }
}

// per-batch keys-only stable LSD radix sort: 8 passes of 4 bits, 1 block/batch
__global__ void __launch_bounds__(256) k_radix_sort(int* bufA, int* bufB) {
  __shared__ unsigned s_hist[16 * 256];
  __shared__ unsigned s_tot[16];
  __shared__ unsigned s_base[16];
  const int tid = threadIdx.x;
  const int b   = blockIdx.x;
  unsigned* A  = (unsigned*)bufA + (size_t)b * kN;
  unsigned* Bp = (unsigned*)bufB + (size_t)b * kN;
  for (int pass = 0; pass < 8; ++pass) {
    unsigned* src = (pass & 1) ? Bp : A;
    unsigned* dst = (pass & 1) ? A  : Bp;
    const int shift = pass * 4;
    for (int i = tid; i < 16 * 256; i += 256) s_hist[i] = 0;
    __syncthreads();
    const int base = tid * 256;              // each thread owns a contiguous chunk
    for (int i = 0; i < 256; ++i) {
      unsigned d = (src[base + i] >> shift) & 15u;
      s_hist[d * 256 + tid]++;
    }
    __syncthreads();
    if (tid < 16) {                          // exclusive scan within digit row
      unsigned run = 0;
      for (int t = 0; t < 256; ++t) {
        unsigned c = s_hist[tid * 256 + t];
        s_hist[tid * 256 + t] = run;
        run += c;
      }
      s_tot[tid] = run;
    }
    __syncthreads();
    if (tid == 0) {                          // digit bases
      unsigned acc = 0;
      for (int d = 0; d < 16; ++d) { s_base[d] = acc; acc += s_tot[d]; }
    }
    __syncthreads();
    for (int i = 0; i < 256; ++i) {          // stable scatter
      unsigned k = src[base + i];
      unsigned d = (k >> shift) & 15u;
      unsigned pos = s_base[d] + s_hist[d * 256 + tid]++;
      dst[pos] = k;
    }
    __syncthreads();
  }
}

// compact sorted keys into unq (unq prefilled with SENT); accumulate valid count
__global__ void __launch_bounds__(256) k_unique(const int* __restrict__ sorted,
                                                int* __restrict__ unq, int* nvalid) {
  __shared__ unsigned s_scan[256];
  __shared__ unsigned s_run;
  const int tid = threadIdx.x;
  const int b   = blockIdx.x;
  const int* s = sorted + (size_t)b * kN;
  int* u = unq + (size_t)b * kN;
  if (tid == 0) s_run = 0;
  __syncthreads();
  for (int chunk = 0; chunk < kN / 256; ++chunk) {
    int i = chunk * 256 + tid;
    int key = s[i];
    unsigned flag = (i == 0) || (key != s[i - 1]);
    s_scan[tid] = flag;
    __syncthreads();
    for (int off = 1; off < 256; off <<= 1) {
      unsigned v = (tid >= off) ? s_scan[tid - off] : 0u;
      __syncthreads();
      s_scan[tid] += v;
      __syncthreads();
    }
    unsigned incl = s_scan[tid];
    unsigned base = s_run;
    if (flag) u[base + incl - 1] = key;
    unsigned tot = s_scan[255];
    __syncthreads();
    if (tid == 0) s_run = base + tot;
    __syncthreads();
  }
  if (tid == 0) atomicAdd(nvalid, (int)s_run);
}

// per point: inv = lower_bound(unq, code); masked feature pooling via atomics
__global__ void k_voxelize(const int* __restrict__ codes, const float* __restrict__ feat,
                           const float* __restrict__ mask, const int* __restrict__ unq,
                           int* __restrict__ inv, float* __restrict__ fsum,
                           float* __restrict__ cnt) {
  int p = blockIdx.x * blockDim.x + threadIdx.x;
  if (p >= kBN) return;
  int b = p / kN;
  int code = codes[p];
  const int* u = unq + (size_t)b * kN;
  int lo = 0, hi = kN;
  while (lo < hi) { int mid = (lo + hi) >> 1; if (u[mid] < code) lo = mid + 1; else hi = mid; }
  inv[p] = lo;
  float m = mask[p];
  if (m != 0.f) {
    int row = b * kN + lo;
    atomicAdd(&cnt[row], m);
    const float* f = feat + (size_t)p * 16;
    float* fs = fsum + (size_t)row * 16;
#pragma unroll
    for (int c = 0; c < 16; ++c) atomicAdd(&fs[c], f[c] * m);
  }
}

// vfeat = fsum / max(cnt, 1)  (in place)
__global__ void k_vfeat(float* __restrict__ fsum, const float* __restrict__ cnt) {
  long i = blockIdx.x * (long)blockDim.x + threadIdx.x;
  if (i >= (long)kBN * 16) return;
  float c = fmaxf(cnt[i >> 4], 1.f);
  fsum[i] = fsum[i] / c;
}

// neighbor table: nbr[g*27+ko] = global feature row of neighbor voxel, or -1
__global__ void k_nbr(const int* __restrict__ unq, int* __restrict__ nbr) {
  int g = blockIdx.x * blockDim.x + threadIdx.x;
  if (g >= kBN) return;
  int code = unq[g];
  int* out = nbr + (size_t)g * 27;
  if (code >= kSENT) {
    for (int ko = 0; ko < 27; ++ko) out[ko] = -1;
    return;
  }
  int b = g / kN;
  const int* u = unq + (size_t)b * kN;
  int x = code % 1000, y = (code / 1000) % 1000, z = code / 1000000;
  for (int ko = 0; ko < 27; ++ko) {
    int dz = ko / 9 - 1, dy = (ko / 3) % 3 - 1, dx = ko % 3 - 1;
    int nx = x + dx, ny = y + dy, nz = z + dz;
    int res = -1;
    if (nx >= 0 && nx < 1000 && ny >= 0 && ny < 1000 && nz >= 0 && nz < 1000) {
      int nc = code + dx + 1000 * dy + 1000000 * dz;
      int lo = 0, hi = kN;
      while (lo < hi) { int mid = (lo + hi) >> 1; if (u[mid] < nc) lo = mid + 1; else hi = mid; }
      if (lo < kN && u[lo] == nc) res = b * kN + lo;
    }
    out[ko] = res;
  }
}

// ---------------- weight swizzle into WMMA B fragments ----------------
// Wt fragment (ko, kb, ct): 32 lanes x float2, contiguous 256B.
// lane L (m=L&15, hf=L>>4): {W[ko][kb*4+2*hf][ct*16+m], W[ko][kb*4+2*hf+1][ct*16+m]}
__global__ void k_wt_swizzle(const float* __restrict__ W, float* __restrict__ Wt,
                             int CIN, int COUT) {
  int total = 27 * CIN * COUT;
  int j = blockIdx.x * blockDim.x + threadIdx.x;
  if (j >= total) return;
  int nCT = COUT >> 4;
  int nKB = CIN >> 2;
  int frag = j >> 6;
  int within = j & 63;
  int lane = within >> 1;
  int e = within & 1;
  int m = lane & 15, hf = lane >> 4;
  int ct = frag % nCT;
  int kb = (frag / nCT) % nKB;
  int ko = frag / (nCT * nKB);
  int kr = kb * 4 + hf * 2 + e;
  int co = ct * 16 + m;
  Wt[j] = W[((size_t)ko * CIN + kr) * COUT + co];
}

// ---------------- 27-tap subm conv via V_WMMA_F32_16X16X4_F32 ----------------
// One wave -> 4 voxel tiles (64 rows) x one 16-cout tile; B fragment loaded
// once per k-step (coalesced b64) and reused for all 4 WMMAs.
template <int CIN>
__global__ void __launch_bounds__(256) k_conv_wmma(const float* __restrict__ feat,
                                                   const float* __restrict__ Wt,
                                                   const float* __restrict__ bias,
                                                   const int* __restrict__ nbr,
                                                   const int* __restrict__ unq,
                                                   float* __restrict__ out, int COUT) {
  const int lane = threadIdx.x & 31;
  const int wave = threadIdx.x >> 5;
  const int job  = blockIdx.x * 8 + wave;    // 4 consecutive 16-row tiles
  const int ct   = blockIdx.y;               // cout tile
  const int nCT  = COUT >> 4;
  const int co0  = ct * 16;
  const int m    = lane & 15;
  const int hf   = lane >> 4;
  const int rowBase = job * 64;
  v8f acc0 = {0.f,0.f,0.f,0.f,0.f,0.f,0.f,0.f};
  v8f acc1 = acc0, acc2 = acc0, acc3 = acc0;
  for (int ko = 0; ko < 27; ++ko) {
    const int nr0 = nbr[(size_t)(rowBase +  0 + m) * 27 + ko];
    const int nr1 = nbr[(size_t)(rowBase + 16 + m) * 27 + ko];
    const int nr2 = nbr[(size_t)(rowBase + 32 + m) * 27 + ko];
    const int nr3 = nbr[(size_t)(rowBase + 48 + m) * 27 + ko];
    const int   r0 = nr0 < 0 ? 0 : nr0;  const float f0 = nr0 < 0 ? 0.f : 1.f;
    const int   r1 = nr1 < 0 ? 0 : nr1;  const float f1 = nr1 < 0 ? 0.f : 1.f;
    const int   r2 = nr2 < 0 ? 0 : nr2;  const float f2 = nr2 < 0 ? 0.f : 1.f;
    const int   r3 = nr3 < 0 ? 0 : nr3;  const float f3 = nr3 < 0 ? 0.f : 1.f;
    const float* fragKo = Wt + ((size_t)ko * (CIN / 4) * nCT + ct) * 64 + lane * 2;
#pragma unroll
    for (int kb = 0; kb < CIN / 4; ++kb) {
      v2f bv = *(const v2f*)(fragKo + (size_t)kb * nCT * 64);
      const int kr = kb * 4 + hf * 2;
      v2f a0 = *(const v2f*)(feat + (size_t)r0 * CIN + kr); a0.x *= f0; a0.y *= f0;
      v2f a1 = *(const v2f*)(feat + (size_t)r1 * CIN + kr); a1.x *= f1; a1.y *= f1;
      v2f a2 = *(const v2f*)(feat + (size_t)r2 * CIN + kr); a2.x *= f2; a2.y *= f2;
      v2f a3 = *(const v2f*)(feat + (size_t)r3 * CIN + kr); a3.x *= f3; a3.y *= f3;
      acc0 = __builtin_amdgcn_wmma_f32_16x16x4_f32(false, a0, false, bv, (short)0, acc0, false, false);
      acc1 = __builtin_amdgcn_wmma_f32_16x16x4_f32(false, a1, false, bv, (short)0, acc1, false, false);
      acc2 = __builtin_amdgcn_wmma_f32_16x16x4_f32(false, a2, false, bv, (short)0, acc2, false, false);
      acc3 = __builtin_amdgcn_wmma_f32_16x16x4_f32(false, a3, false, bv, (short)0, acc3, false, false);
    }
  }
  const float bval = bias[co0 + m];
#pragma unroll
  for (int t = 0; t < 4; ++t) {
    const v8f& acc = (t == 0) ? acc0 : (t == 1) ? acc1 : (t == 2) ? acc2 : acc3;
#pragma unroll
    for (int r = 0; r < 8; ++r) {
      const int row = rowBase + t * 16 + r + hf * 8;
      const float validf = (unq[row] < kSENT) ? 1.f : 0.f;
      out[(size_t)row * COUT + co0 + m] = (acc[r] + bval) * validf;
    }
  }
}

// ---------------- batch norm ----------------
__global__ void k_zero_stats(float* sum, float* sq) {
  if (threadIdx.x < 128) { sum[threadIdx.x] = 0.f; sq[threadIdx.x] = 0.f; }
}

__global__ void k_bn_reduce(const float* __restrict__ x, float* __restrict__ sum,
                            float* __restrict__ sq, int C) {
  __shared__ float ssum[128];
  __shared__ float ssq[128];
  for (int c = threadIdx.x; c < C; c += blockDim.x) { ssum[c] = 0.f; ssq[c] = 0.f; }
  __syncthreads();
  const long total = (long)kBN * C;
  for (long i = blockIdx.x * (long)blockDim.x + threadIdx.x; i < total;
       i += (long)gridDim.x * blockDim.x) {
    float v = x[i];
    int c = (int)(i & (C - 1));              // C is a power of two
    atomicAdd(&ssum[c], v);
    atomicAdd(&ssq[c], v * v);
  }
  __syncthreads();
  for (int c = threadIdx.x; c < C; c += blockDim.x) {
    atomicAdd(&sum[c], ssum[c]);
    atomicAdd(&sq[c], ssq[c]);
  }
}

__global__ void k_bn_apply(float* __restrict__ x, const float* __restrict__ sum,
                           const float* __restrict__ sq, const float* __restrict__ g,
                           const float* __restrict__ be, const int* __restrict__ unq,
                           const int* __restrict__ nvalid, int C) {
  const float n = fmaxf((float)(*nvalid), 1.f);
  const long total = (long)kBN * C;
  for (long i = blockIdx.x * (long)blockDim.x + threadIdx.x; i < total;
       i += (long)gridDim.x * blockDim.x) {
    int c = (int)(i & (C - 1));
    long row = i / C;
    float mean = sum[c] / n;
    float var  = fmaxf(sq[c] / n - mean * mean, 0.f);
    float v = (x[i] - mean) * rsqrtf(var + kEPS) * g[c] + be[c];
    v = fmaxf(v, 0.f);
    x[i] = (unq[row] < kSENT) ? v : 0.f;
  }
}

// ---------------- gather voxel features back to points ----------------
__global__ void k_gather(const float* __restrict__ h3, const int* __restrict__ inv,
                         const float* __restrict__ mask, float* __restrict__ out) {
  long i = blockIdx.x * (long)blockDim.x + threadIdx.x;   // one float4 per thread
  if (i >= (long)kBN * 32) return;
  int p = (int)(i >> 5);
  int q = (int)(i & 31) * 4;
  int b = p / kN;
  float m = mask[p];
  const float4 s = *(const float4*)(h3 + ((size_t)(b * kN + inv[p])) * 128 + q);
  float4 d; d.x = s.x * m; d.y = s.y * m; d.z = s.z * m; d.w = s.w * m;
  *(float4*)(out + (size_t)p * 128 + q) = d;
}

// ---------------------------------------------------------------------------
extern "C" void kernel_launch(void* const* d_in, const int* in_sizes, int n_in,
                              void* d_out, int out_size, void* d_ws, size_t ws_size,
                              hipStream_t stream) {
  (void)in_sizes; (void)n_in; (void)out_size; (void)ws_size;
  const float* xyz  = (const float*)d_in[0];
  const float* feat = (const float*)d_in[1];
  const float* mask = (const float*)d_in[2];
  const float* W1  = (const float*)d_in[3];
  const float* b1  = (const float*)d_in[4];
  const float* g1  = (const float*)d_in[5];
  const float* be1 = (const float*)d_in[6];
  const float* W2  = (const float*)d_in[7];
  const float* b2  = (const float*)d_in[8];
  const float* g2  = (const float*)d_in[9];
  const float* be2 = (const float*)d_in[10];
  const float* W3  = (const float*)d_in[11];
  const float* b3  = (const float*)d_in[12];
  const float* g3  = (const float*)d_in[13];
  const float* be3 = (const float*)d_in[14];

  char* ws = (char*)d_ws;
  unsigned* origin = (unsigned*)(ws + OFF_ORIGIN);
  int*    nvalid = (int*)(ws + OFF_NVALID);
  float*  bnsum  = (float*)(ws + OFF_BNSUM);
  float*  bnsq   = (float*)(ws + OFF_BNSQ);
  float*  Wt     = (float*)(ws + OFF_WT);
  int*    codes  = (int*)(ws + OFF_CODES);
  int*    sortA  = (int*)(ws + OFF_SORTA);
  int*    sortB  = (int*)(ws + OFF_SORTB);
  int*    unq    = (int*)(ws + OFF_UNQ);
  int*    inv    = (int*)(ws + OFF_INV);
  float*  cnt    = (float*)(ws + OFF_CNT);
  int*    nbr    = (int*)(ws + OFF_NBR);
  float*  vfeat  = (float*)(ws + OFF_VFEAT);
  float*  h1     = (float*)(ws + OFF_H1);
  float*  h2     = (float*)(ws + OFF_H2);
  float*  h3     = (float*)(ws + OFF_H3);
  float*  out    = (float*)d_out;

  // ---- voxelization pipeline ----
  k_init<<<1, 64, 0, stream>>>(origin, nvalid);
  k_min_xyz<<<256, 256, 0, stream>>>(xyz, origin);
  k_codes<<<kBN / 256, 256, 0, stream>>>(xyz, origin, codes, sortA);
  k_prep<<<(kBN * 16) / 256, 256, 0, stream>>>(unq, cnt, vfeat);
  k_radix_sort<<<kB, 256, 0, stream>>>(sortA, sortB);
  k_unique<<<kB, 256, 0, stream>>>(sortA, unq, nvalid);
  k_voxelize<<<kBN / 256, 256, 0, stream>>>(codes, feat, mask, unq, inv, vfeat, cnt);
  k_vfeat<<<(kBN * 16) / 256, 256, 0, stream>>>(vfeat, cnt);
  k_nbr<<<kBN / 256, 256, 0, stream>>>(unq, nbr);

  const int jobBlocks = kBN / 64 / 8;  // 8 waves/block, 64 rows per wave

  // ---- layer 1: 16 -> 32 ----
  k_wt_swizzle<<<(27 * 16 * 32 + 255) / 256, 256, 0, stream>>>(W1, Wt, 16, 32);
  k_zero_stats<<<1, 128, 0, stream>>>(bnsum, bnsq);
  k_conv_wmma<16><<<dim3(jobBlocks, 2), 256, 0, stream>>>(vfeat, Wt, b1, nbr, unq, h1, 32);
  k_bn_reduce<<<2048, 256, 0, stream>>>(h1, bnsum, bnsq, 32);
  k_bn_apply<<<4096, 256, 0, stream>>>(h1, bnsum, bnsq, g1, be1, unq, nvalid, 32);

  // ---- layer 2: 32 -> 64 ----
  k_wt_swizzle<<<(27 * 32 * 64 + 255) / 256, 256, 0, stream>>>(W2, Wt, 32, 64);
  k_zero_stats<<<1, 128, 0, stream>>>(bnsum, bnsq);
  k_conv_wmma<32><<<dim3(jobBlocks, 4), 256, 0, stream>>>(h1, Wt, b2, nbr, unq, h2, 64);
  k_bn_reduce<<<2048, 256, 0, stream>>>(h2, bnsum, bnsq, 64);
  k_bn_apply<<<4096, 256, 0, stream>>>(h2, bnsum, bnsq, g2, be2, unq, nvalid, 64);

  // ---- layer 3: 64 -> 128 ----
  k_wt_swizzle<<<(27 * 64 * 128 + 255) / 256, 256, 0, stream>>>(W3, Wt, 64, 128);
  k_zero_stats<<<1, 128, 0, stream>>>(bnsum, bnsq);
  k_conv_wmma<64><<<dim3(jobBlocks, 8), 256, 0, stream>>>(h2, Wt, b3, nbr, unq, h3, 128);
  k_bn_reduce<<<2048, 256, 0, stream>>>(h3, bnsum, bnsq, 128);
  k_bn_apply<<<4096, 256, 0, stream>>>(h3, bnsum, bnsq, g3, be3, unq, nvalid, 128);

  // ---- scatter back to points ----
  k_gather<<<(kBN * 32) / 256, 256, 0, stream>>>(h3, inv, mask, out);
}